// V_Max_89275190215347
// MI455X (gfx1250) — compile-verified
//
#include <hip/hip_runtime.h>
#include <stdint.h>

// Problem constants from the reference: B=8, T=12, N=10000, D=16, E=160000.
#define NNODES 10000
#define NROWS  96      // B*T
#define DDIM   16
#define EPB    64      // edges per block (8 waves x 8 edges)

// ---------------------------------------------------------------------------
// Pass 1: init output to -inf so that per-destination max is exact even when
// every incoming message is negative.
// ---------------------------------------------------------------------------
__global__ __launch_bounds__(256) void vmax_init(float4* __restrict__ out, int n4) {
  int i = blockIdx.x * blockDim.x + threadIdx.x;
  if (i >= n4) return;
  float4 v;
  v.x = v.y = v.z = v.w = -__builtin_inff();
  out[i] = v;
}

// ---------------------------------------------------------------------------
// Pass 2: edge-parallel scatter-max.
//   - One wave32 per edge. Lane layout: lnHi = lane>>2 picks one of 8 (b,t)
//     rows per iteration, lnLo = lane&3 picks a float4 within the 16-float row.
//   - 12 iterations cover all 96 (b,t) rows: 1 x global_load_b128 (coalesced,
//     64B per row) + 4 x returnless global_atomic_max_num_f32 at DEV scope.
//   - Edge metadata staged to LDS via the CDNA5 async-to-LDS path (ASYNCcnt).
// ---------------------------------------------------------------------------
__global__ __launch_bounds__(256) void vmax_scatter(
    const float* __restrict__ V,
    const int*   __restrict__ src,
    const int*   __restrict__ dst,
    const float* __restrict__ w,
    float*       __restrict__ out,
    int E)
{
  __shared__ int   sSrc[EPB];
  __shared__ int   sDst[EPB];
  __shared__ float sW[EPB];

  const int tid       = threadIdx.x;
  const int blockBase = blockIdx.x * EPB;

  // --- async-stage this block's edge data into LDS (gfx1250 ASYNCcnt path) ---
  if (tid < EPB) {
    if (blockBase + tid < E) {
      unsigned lds = (unsigned)(uintptr_t)&sSrc[tid];
      unsigned go  = (unsigned)((blockBase + tid) * 4u);
      asm volatile("global_load_async_to_lds_b32 %0, %1, %2"
                   :: "v"(lds), "v"(go), "s"((uint64_t)(uintptr_t)src) : "memory");
    }
  } else if (tid < 2 * EPB) {
    if (blockBase + (tid - EPB) < E) {
      unsigned lds = (unsigned)(uintptr_t)&sDst[tid - EPB];
      unsigned go  = (unsigned)((blockBase + tid - EPB) * 4u);
      asm volatile("global_load_async_to_lds_b32 %0, %1, %2"
                   :: "v"(lds), "v"(go), "s"((uint64_t)(uintptr_t)dst) : "memory");
    }
  } else if (tid < 3 * EPB) {
    if (blockBase + (tid - 2 * EPB) < E) {
      unsigned lds = (unsigned)(uintptr_t)&sW[tid - 2 * EPB];
      unsigned go  = (unsigned)((blockBase + tid - 2 * EPB) * 4u);
      asm volatile("global_load_async_to_lds_b32 %0, %1, %2"
                   :: "v"(lds), "v"(go), "s"((uint64_t)(uintptr_t)w) : "memory");
    }
  }
  asm volatile("s_wait_asynccnt 0x0" ::: "memory");
  __syncthreads();

  const int wave = tid >> 5;
  const int lane = tid & 31;
  const int lnHi = lane >> 2;   // row within the 8-row iteration group
  const int lnLo = lane & 3;    // float4 slot within the 16-float row

  const char*    vbase = (const char*)V;
  const uint64_t obase = (uint64_t)(uintptr_t)out;
  const unsigned stride = 8u * NNODES * DDIM * 4u;  // byte stride per 8-row step

  for (int k = 0; k < EPB / 8; ++k) {
    const int eLocal = (wave << 3) + k;   // wave-uniform
    const int e      = blockBase + eLocal;
    if (e >= E) continue;

    const int   s  = sSrc[eLocal];
    const int   d  = sDst[eLocal];
    const float wt = sW[eLocal];

    // byte offsets for row r = lnHi, float4 column lnLo
    unsigned vOff = ((unsigned)(lnHi * NNODES + s) * (DDIM * 4u)) + (unsigned)(lnLo * 16);
    unsigned oOff = ((unsigned)(lnHi * NNODES + d) * (DDIM * 4u)) + (unsigned)(lnLo * 16);

#pragma unroll
    for (int it = 0; it < NROWS / 8; ++it) {
      const float4 v = *(const float4*)(vbase + vOff);
      float4 m;
      m.x = v.x * wt; m.y = v.y * wt; m.z = v.z * wt; m.w = v.w * wt;
      // Returnless f32 max atomics, device scope (cross-WGP correctness).
      asm volatile("global_atomic_max_num_f32 %0, %1, %2 scope:SCOPE_DEV"
                   :: "v"(oOff), "v"(m.x), "s"(obase) : "memory");
      asm volatile("global_atomic_max_num_f32 %0, %1, %2 offset:4 scope:SCOPE_DEV"
                   :: "v"(oOff), "v"(m.y), "s"(obase) : "memory");
      asm volatile("global_atomic_max_num_f32 %0, %1, %2 offset:8 scope:SCOPE_DEV"
                   :: "v"(oOff), "v"(m.z), "s"(obase) : "memory");
      asm volatile("global_atomic_max_num_f32 %0, %1, %2 offset:12 scope:SCOPE_DEV"
                   :: "v"(oOff), "v"(m.w), "s"(obase) : "memory");
      vOff += stride;
      oOff += stride;
    }
  }
}

// ---------------------------------------------------------------------------
// Pass 3: nodes with no incoming edges still hold -inf -> replace with 0
// (reference: where(isneginf(out), 0, out)).
// ---------------------------------------------------------------------------
__global__ __launch_bounds__(256) void vmax_fixup(float4* __restrict__ out, int n4) {
  int i = blockIdx.x * blockDim.x + threadIdx.x;
  if (i >= n4) return;
  float4 v = out[i];
  const float ninf = -__builtin_inff();
  v.x = (v.x == ninf) ? 0.0f : v.x;
  v.y = (v.y == ninf) ? 0.0f : v.y;
  v.z = (v.z == ninf) ? 0.0f : v.z;
  v.w = (v.w == ninf) ? 0.0f : v.w;
  out[i] = v;
}

extern "C" void kernel_launch(void* const* d_in, const int* in_sizes, int n_in,
                              void* d_out, int out_size, void* d_ws, size_t ws_size,
                              hipStream_t stream) {
  const float* V   = (const float*)d_in[0];
  const int*   src = (const int*)d_in[1];
  const int*   dst = (const int*)d_in[2];
  const float* w   = (const float*)d_in[3];
  // d_in[4] (V_in) and d_in[5] (adp) are unused by the reference computation.
  float* out = (float*)d_out;

  const int E  = in_sizes[1];
  const int n4 = out_size / 4;                 // out_size = B*T*N*D, divisible by 4

  const int initBlocks = (n4 + 255) / 256;
  vmax_init<<<initBlocks, 256, 0, stream>>>((float4*)out, n4);

  const int scatterBlocks = (E + EPB - 1) / EPB;
  vmax_scatter<<<scatterBlocks, 256, 0, stream>>>(V, src, dst, w, out, E);

  vmax_fixup<<<initBlocks, 256, 0, stream>>>((float4*)out, n4);
}